// GraphSAGEClassifier_81879256531434
// MI455X (gfx1250) — compile-verified
//
#include <hip/hip_runtime.h>
#include <hip/hip_bf16.h>

typedef __attribute__((ext_vector_type(2))) float v2f;
typedef __attribute__((ext_vector_type(8))) float v8f;

// ---------------------------------------------------------------------------
// Y = X @ [Wa ; Wb]^T  (row-major, ldy = 2*Nhalf), fp32 via V_WMMA_F32_16X16X4_F32.
// grid.x = ceil(M/32): each block owns a 32-row stripe; each wave owns a
// 32x64 output tile = 2 m-tiles x 4 n-tiles = 8 f32 accumulators.
// Fragment layout (ISA 7.12.2): A 16x4 and B 4x16 are 2 VGPRs each;
// lanes 0-15 hold K={k0,k0+1}, lanes 16-31 hold K={k0+2,k0+3} -> one float2 load
// per lane from X[m,:] and W[n,:] (B[k][n] == W[n][k], so X@W^T is direct).
// ---------------------------------------------------------------------------
__global__ void __launch_bounds__(256)
gemm_xwt_wmma(const float* __restrict__ X,
              const float* __restrict__ Wa,
              const float* __restrict__ Wb,
              float* __restrict__ Y,
              int M, int K, int Nhalf)
{
    const int wave = threadIdx.x >> 5;
    const int lane = threadIdx.x & 31;
    const int half = lane >> 4;      // which K-pair this lane holds
    const int lm   = lane & 15;

    const int mBase = blockIdx.x * 32;   // two 16-row tiles per wave
    const int nTile = wave * 64;         // output column base for this wave
    const int ldy   = 2 * Nhalf;

    const float* Wp;
    int wrow;
    if (nTile < Nhalf) { Wp = Wa; wrow = nTile; }
    else               { Wp = Wb; wrow = nTile - Nhalf; }

    int arow0 = mBase + lm;        if (arow0 >= M) arow0 = M - 1;
    int arow1 = mBase + 16 + lm;   if (arow1 >= M) arow1 = M - 1;

    const float* X0 = X  + (size_t)arow0 * K + 2 * half;
    const float* X1 = X  + (size_t)arow1 * K + 2 * half;
    const float* B0 = Wp + (size_t)(wrow +  0 + lm) * K + 2 * half;
    const float* B1 = Wp + (size_t)(wrow + 16 + lm) * K + 2 * half;
    const float* B2 = Wp + (size_t)(wrow + 32 + lm) * K + 2 * half;
    const float* B3 = Wp + (size_t)(wrow + 48 + lm) * K + 2 * half;

    v8f a00 = {}, a01 = {}, a02 = {}, a03 = {};   // m-tile 0, n-tiles 0..3
    v8f a10 = {}, a11 = {}, a12 = {}, a13 = {};   // m-tile 1, n-tiles 0..3

#pragma unroll 4
    for (int k = 0; k < K; k += 4) {
        v2f a0 = *(const v2f*)(X0 + k);
        v2f a1 = *(const v2f*)(X1 + k);
        v2f b0 = *(const v2f*)(B0 + k);
        v2f b1 = *(const v2f*)(B1 + k);
        v2f b2 = *(const v2f*)(B2 + k);
        v2f b3 = *(const v2f*)(B3 + k);
        a00 = __builtin_amdgcn_wmma_f32_16x16x4_f32(false, a0, false, b0, (short)0, a00, false, false);
        a01 = __builtin_amdgcn_wmma_f32_16x16x4_f32(false, a0, false, b1, (short)0, a01, false, false);
        a02 = __builtin_amdgcn_wmma_f32_16x16x4_f32(false, a0, false, b2, (short)0, a02, false, false);
        a03 = __builtin_amdgcn_wmma_f32_16x16x4_f32(false, a0, false, b3, (short)0, a03, false, false);
        a10 = __builtin_amdgcn_wmma_f32_16x16x4_f32(false, a1, false, b0, (short)0, a10, false, false);
        a11 = __builtin_amdgcn_wmma_f32_16x16x4_f32(false, a1, false, b1, (short)0, a11, false, false);
        a12 = __builtin_amdgcn_wmma_f32_16x16x4_f32(false, a1, false, b2, (short)0, a12, false, false);
        a13 = __builtin_amdgcn_wmma_f32_16x16x4_f32(false, a1, false, b3, (short)0, a13, false, false);
    }

    // C/D layout: VGPR r, lanes 0-15 -> row r, lanes 16-31 -> row r+8; col = lm
#pragma unroll
    for (int r = 0; r < 8; ++r) {
        int m0 = mBase + r + 8 * half;
        if (m0 < M) {
            float* row = Y + (size_t)m0 * ldy + nTile + lm;
            row[0]  = a00[r];
            row[16] = a01[r];
            row[32] = a02[r];
            row[48] = a03[r];
        }
        int m1 = mBase + 16 + r + 8 * half;
        if (m1 < M) {
            float* row = Y + (size_t)m1 * ldy + nTile + lm;
            row[0]  = a10[r];
            row[16] = a11[r];
            row[32] = a12[r];
            row[48] = a13[r];
        }
    }
}

// ---------------------------------------------------------------------------
__global__ void deg_count_kernel(const int* __restrict__ dst,
                                 float* __restrict__ deg, int E)
{
    int e = blockIdx.x * blockDim.x + threadIdx.x;
    if (e < E) atomicAdd(&deg[dst[e]], 1.0f);
}

// agg[dst[e], :] += feat[src[e], 0:C]   (feat row stride ldf)
__global__ void edge_scatter_kernel(const float* __restrict__ feat, int ldf,
                                    const int* __restrict__ src,
                                    const int* __restrict__ dst,
                                    float* __restrict__ agg, int C, int E)
{
    long long idx = (long long)blockIdx.x * blockDim.x + threadIdx.x;
    int cpe = C >> 2;                       // float4 chunks per edge
    long long e = idx / cpe;
    int c4 = (int)(idx % cpe) << 2;
    if (e >= E) return;
    int s = src[e], d = dst[e];
    const float4 v = *(const float4*)(feat + (size_t)s * ldf + c4);
    float* o = agg + (size_t)d * C + c4;
    atomicAdd(o + 0, v.x);
    atomicAdd(o + 1, v.y);
    atomicAdd(o + 2, v.z);
    atomicAdd(o + 3, v.w);
}

// agg[i,c] = relu(agg[i,c]/max(deg[i],1) + bias[c] + other[i, ocol0+c])
__global__ void mean_bias_relu_kernel(float* __restrict__ agg,
                                      const float* __restrict__ deg,
                                      const float* __restrict__ bias,
                                      const float* __restrict__ other,
                                      int ldo, int ocol0, int C, int Nn)
{
    long long idx = (long long)blockIdx.x * blockDim.x + threadIdx.x;
    long long i = idx / C;
    int c = (int)(idx % C);
    if (i >= Nn) return;
    float dg = deg[i]; dg = dg > 1.0f ? dg : 1.0f;
    float v = agg[(size_t)i * C + c] / dg + bias[c]
            + other[(size_t)i * ldo + ocol0 + c];
    agg[(size_t)i * C + c] = v > 0.0f ? v : 0.0f;
}

// layer-2 epilogue fused with graph mean-pool accumulation
__global__ void mean_bias_relu_pool_kernel(const float* __restrict__ agg,
                                           const float* __restrict__ deg,
                                           const float* __restrict__ bias,
                                           const float* __restrict__ other,
                                           int ldo, int ocol0,
                                           const int* __restrict__ batch,
                                           float* __restrict__ pooled,
                                           float* __restrict__ cnt,
                                           int C, int Nn)
{
    long long idx = (long long)blockIdx.x * blockDim.x + threadIdx.x;
    long long i = idx / C;
    int c = (int)(idx % C);
    if (i >= Nn) return;
    float dg = deg[i]; dg = dg > 1.0f ? dg : 1.0f;
    float v = agg[(size_t)i * C + c] / dg + bias[c]
            + other[(size_t)i * ldo + ocol0 + c];
    v = v > 0.0f ? v : 0.0f;
    int g = batch[i];
    atomicAdd(&pooled[(size_t)g * C + c], v);
    if (c == 0) atomicAdd(&cnt[g], 1.0f);
}

// out[g,k] = bc[k] + (1/max(cnt[g],1)) * sum_c pooled[g,c]*Wc[k,c]
__global__ void classifier_kernel(const float* __restrict__ pooled,
                                  const float* __restrict__ cnt,
                                  const float* __restrict__ Wc,
                                  const float* __restrict__ bc,
                                  float* __restrict__ out, int C, int NC)
{
    int g = blockIdx.x;
    int k = threadIdx.x;
    if (k >= NC) return;
    float cg = cnt[g]; cg = cg > 1.0f ? cg : 1.0f;
    const float* p = pooled + (size_t)g * C;
    const float* w = Wc + (size_t)k * C;
    float acc = 0.0f;
    for (int c = 0; c < C; ++c) acc += p[c] * w[c];
    out[g * NC + k] = bc[k] + acc / cg;
}

// ---------------------------------------------------------------------------
extern "C" void kernel_launch(void* const* d_in, const int* in_sizes, int n_in,
                              void* d_out, int out_size, void* d_ws, size_t ws_size,
                              hipStream_t stream)
{
    const float* x    = (const float*)d_in[0];
    const int*   ei   = (const int*)d_in[1];   // [2, E] int32
    const int*   batch= (const int*)d_in[2];
    const float* W1l  = (const float*)d_in[3];
    const float* b1l  = (const float*)d_in[4];
    const float* W1r  = (const float*)d_in[5];
    const float* W2l  = (const float*)d_in[6];
    const float* b2l  = (const float*)d_in[7];
    const float* W2r  = (const float*)d_in[8];
    const float* Wc   = (const float*)d_in[9];
    const float* bc   = (const float*)d_in[10];

    const int IN = 1024, HID = 256, EMB = 128, NC = 16, NG = 64;
    const int Nn = in_sizes[0] / IN;
    const int E  = in_sizes[1] / 2;
    const int* src = ei;
    const int* dst = ei + E;

    // workspace layout (floats)
    float* buf0   = (float*)d_ws;                      // [N, 512] gemm outputs
    float* buf1   = buf0 + (size_t)Nn * 512;           // [N, 256] agg / hidden
    float* deg    = buf1 + (size_t)Nn * 256;           // [N]
    float* pooled = deg + Nn;                          // [NG, EMB]
    float* cnt    = pooled + (size_t)NG * EMB;         // [NG]

    const int mt = (Nn + 31) / 32;   // 32 rows per block

    // ---- layer 1: [lin_l(x) | lin_r(x)] -> buf0 [N, 512] ----
    gemm_xwt_wmma<<<mt, 256, 0, stream>>>(x, W1l, W1r, buf0, Nn, IN, HID);

    hipMemsetAsync(buf1, 0, (size_t)Nn * HID * sizeof(float), stream);
    hipMemsetAsync(deg,  0, (size_t)Nn * sizeof(float), stream);
    deg_count_kernel<<<(E + 255) / 256, 256, 0, stream>>>(dst, deg, E);
    {
        long long tot = (long long)E * (HID / 4);
        edge_scatter_kernel<<<(unsigned)((tot + 255) / 256), 256, 0, stream>>>(
            buf0, 512, src, dst, buf1, HID, E);
    }
    {
        long long tot = (long long)Nn * HID;
        mean_bias_relu_kernel<<<(unsigned)((tot + 255) / 256), 256, 0, stream>>>(
            buf1, deg, b1l, buf0, 512, HID, HID, Nn);
    }

    // ---- layer 2: [lin_l(h) | lin_r(h)] -> buf0 [N, 256] ----
    gemm_xwt_wmma<<<mt, 128, 0, stream>>>(buf1, W2l, W2r, buf0, Nn, HID, EMB);

    hipMemsetAsync(buf1, 0, (size_t)Nn * EMB * sizeof(float), stream);
    {
        long long tot = (long long)E * (EMB / 4);
        edge_scatter_kernel<<<(unsigned)((tot + 255) / 256), 256, 0, stream>>>(
            buf0, 256, src, dst, buf1, EMB, E);
    }

    hipMemsetAsync(pooled, 0, (size_t)NG * EMB * sizeof(float), stream);
    hipMemsetAsync(cnt,    0, (size_t)NG * sizeof(float), stream);
    {
        long long tot = (long long)Nn * EMB;
        mean_bias_relu_pool_kernel<<<(unsigned)((tot + 255) / 256), 256, 0, stream>>>(
            buf1, deg, b2l, buf0, 256, EMB, batch, pooled, cnt, EMB, Nn);
    }

    classifier_kernel<<<NG, 32, 0, stream>>>(pooled, cnt, Wc, bc, (float*)d_out, EMB, NC);
}